// QuantumDeepField_52201032515889
// MI455X (gfx1250) — compile-verified
//
#include <hip/hip_runtime.h>
#include <hip/hip_bf16.h>
#include <math.h>

// ---- problem constants (match reference) ----
#define NB   32
#define NF   2000
#define NO   60
#define NDIM 250
#define NDP  256          // padded DIM (zero-filled columns 250..255)
#define NL   3
#define NBO  (NB * NO)    // 1920
#define NBF  (NB * NF)    // 64000
#define EPSN    1e-12f
#define BN_EPS  1e-5f

#define LDSW 272          // LDS row stride (floats): 2*272 mod 64 == 32 -> halves hit disjoint banks
#define LAYER_LDS_BYTES (NDP * LDSW * 4)   // 278528 B <= 320KB WGP LDS

typedef __attribute__((ext_vector_type(2))) float v2f;
typedef __attribute__((ext_vector_type(8))) float v8f;

// D = A(16x4 f32) * B(4x16 f32) + C(16x16 f32), wave32 fp32 matrix pipe.
__device__ __forceinline__ v8f wmma4(v2f a, v2f b, v8f c) {
    return __builtin_amdgcn_wmma_f32_16x16x4_f32(
        false, a, false, b, (short)0, c, false, false);
}

// A fragment: row-major A[* x lda], tile origin (m0,k0). One b64 load per lane.
__device__ __forceinline__ v2f load_a2(const float* __restrict__ A, int lda,
                                       int m0, int k0, int lane) {
    const int r  = m0 + (lane & 15);
    const int kk = k0 + ((lane >> 4) << 1);
    return *(const v2f*)(A + (size_t)r * lda + kk);
}

// B fragment from global: row-major B[* x ldb], tile origin (k0,n0).
__device__ __forceinline__ v2f load_b2(const float* __restrict__ Bp, int ldb,
                                       int k0, int n0, int lane) {
    const int n  = n0 + (lane & 15);
    const int kk = k0 + ((lane >> 4) << 1);
    v2f b;
    b.x = Bp[(size_t)kk * ldb + n];
    b.y = Bp[(size_t)(kk + 1) * ldb + n];
    return b;
}

// C/D store: VGPR r holds M=m0+r (lanes 0-15) / M=m0+r+8 (lanes 16-31), N=n0+(lane&15).
__device__ __forceinline__ void store_c(float* __restrict__ C, int ldc,
                                        int m0, int n0, int lane, v8f acc) {
    const int n  = n0 + (lane & 15);
    const int mh = (lane >> 4) * 8;
#pragma unroll
    for (int r = 0; r < 8; ++r)
        C[(size_t)(m0 + mh + r) * ldc + n] = acc[r];
}

// ---------------------------------------------------------------------------
// Prep: WfT[l][k*NDP+n] = Wf[l][n*NDIM+k] (zero-padded transpose).
__global__ void prep_wft(const float* __restrict__ Wf, float* __restrict__ WfT) {
    int idx = blockIdx.x * blockDim.x + threadIdx.x;
    if (idx >= NL * NDP * NDP) return;
    int l = idx / (NDP * NDP);
    int rem = idx % (NDP * NDP);
    int k = rem / NDP, n = rem % NDP;
    float v = 0.0f;
    if (k < NDIM && n < NDIM)
        v = Wf[(size_t)l * NDIM * NDIM + (size_t)n * NDIM + k];
    WfT[idx] = v;
}

// Prep: pad [rows, NDIM] -> [rows, NDP] with zeros.
__global__ void pad_cols(const float* __restrict__ src, float* __restrict__ dst, int rows) {
    int idx = blockIdx.x * blockDim.x + threadIdx.x;
    if (idx >= rows * NDP) return;
    int r = idx / NDP, c = idx % NDP;
    dst[idx] = (c < NDIM) ? src[r * NDIM + c] : 0.0f;
}

// ---------------------------------------------------------------------------
// K1: block-diagonal radials + per-column (over F) L2 norm scale. grid (NO,NB), block 256.
__global__ void radial_kernel(const float* __restrict__ dist, const int* __restrict__ qn,
                              const int* __restrict__ ao, const float* __restrict__ zeta_emb,
                              float* __restrict__ rad, float* __restrict__ colscale) {
    const int o = blockIdx.x, b = blockIdx.y;
    const int n = qn[b * NO + o];                 // 1,2,3
    const float zeta = zeta_emb[ao[b * NO + o]];
    float ss = 0.0f;
    for (int f = threadIdx.x; f < NF; f += blockDim.x) {
        float d = dist[((size_t)b * NF + f) * NO + o];
        float p = (n == 1) ? 1.0f : ((n == 2) ? d : d * d);   // d^(n-1)
        float r = p * expf(-zeta * d * d);
        rad[((size_t)b * NF + f) * NO + o] = r;
        ss += r * r;
    }
    __shared__ float red[256];
    red[threadIdx.x] = ss;
    __syncthreads();
    for (int s = 128; s > 0; s >>= 1) {
        if (threadIdx.x < s) red[threadIdx.x] += red[threadIdx.x + s];
        __syncthreads();
    }
    if (threadIdx.x == 0)
        colscale[b * NO + o] = 1.0f / fmaxf(sqrtf(red[0]), EPSN);
}

// K2: column-normalized embeddings into padded [NBO, NDP]. grid NB, block 256 (== NDP).
__global__ void coef_kernel(const int* __restrict__ ao, const float* __restrict__ emb,
                            float* __restrict__ coefsN) {
    const int b = blockIdx.x;
    __shared__ int aos[NO];
    if (threadIdx.x < NO) aos[threadIdx.x] = ao[b * NO + threadIdx.x];
    __syncthreads();
    const int d = threadIdx.x;
    if (d < NDIM) {
        float ss = 0.0f;
        for (int o = 0; o < NO; ++o) {
            float e = emb[(size_t)aos[o] * NDIM + d];
            ss += e * e;
        }
        float s = 1.0f / fmaxf(sqrtf(ss), EPSN);
        for (int o = 0; o < NO; ++o)
            coefsN[((size_t)b * NO + o) * NDP + d] = emb[(size_t)aos[o] * NDIM + d] * s;
    } else {
        for (int o = 0; o < NO; ++o)
            coefsN[((size_t)b * NO + o) * NDP + d] = 0.0f;
    }
}

// K3: W[1920,NDP] = Y_lm[1920,1920] @ coefsN[1920,NDP]. grid (NDP/32, NBO/16), block 32.
__global__ void gemm_Ycoef(const float* __restrict__ Ylm, const float* __restrict__ coefsN,
                           float* __restrict__ W) {
    const int lane = threadIdx.x;
    const int n0 = blockIdx.x * 32;
    const int m0 = blockIdx.y * 16;
    v8f acc0 = {}, acc1 = {};
#pragma unroll 4
    for (int k = 0; k < NBO; k += 4) {
        v2f a  = load_a2(Ylm, NBO, m0, k, lane);
        v2f b0 = load_b2(coefsN, NDP, k, n0, lane);
        v2f b1 = load_b2(coefsN, NDP, k, n0 + 16, lane);
        acc0 = wmma4(a, b0, acc0);
        acc1 = wmma4(a, b1, acc1);
    }
    store_c(W, NDP, m0, n0, lane, acc0);
    store_c(W, NDP, m0, n0 + 16, lane, acc1);
}

// K4: per molecule b: MOs[b] = (rad[b]*colscale[b])[2000,60] @ W[b*O:(b+1)*O, NDP].
// grid (NDP/32, NF/16, NB), block 32. K=60 is a multiple of 4; no guards needed.
__global__ void gemm_mos(const float* __restrict__ rad, const float* __restrict__ colscale,
                         const float* __restrict__ W, float* __restrict__ mos) {
    const int lane = threadIdx.x;
    const int n0 = blockIdx.x * 32;
    const int m0 = blockIdx.y * 16;
    const int b  = blockIdx.z;
    const float* A  = rad + (size_t)b * NF * NO;
    const float* Bp = W   + (size_t)b * NO * NDP;
    const float* cs = colscale + b * NO;
    v8f acc0 = {}, acc1 = {};
#pragma unroll 5
    for (int k = 0; k < NO; k += 4) {
        const int kk = k + ((lane >> 4) << 1);
        v2f a = *(const v2f*)(A + (size_t)(m0 + (lane & 15)) * NO + kk);
        a.x *= cs[kk];
        a.y *= cs[kk + 1];
        v2f b0 = load_b2(Bp, NDP, k, n0, lane);
        v2f b1 = load_b2(Bp, NDP, k, n0 + 16, lane);
        acc0 = wmma4(a, b0, acc0);
        acc1 = wmma4(a, b1, acc1);
    }
    float* Cout = mos + (size_t)b * NF * NDP;
    store_c(Cout, NDP, m0, n0, lane, acc0);
    store_c(Cout, NDP, m0, n0 + 16, lane, acc1);
}

// K5: mosnorm[b,d] = sqrt(Ne[b]/DIM) / max(||MOs[b,:,d]||, eps); pad cols -> 0.
// grid (NDP, NB), block 256.
__global__ void mosnorm_kernel(const float* __restrict__ mos, const float* __restrict__ Ne,
                               float* __restrict__ mosnorm) {
    const int d = blockIdx.x, b = blockIdx.y;
    if (d >= NDIM) {
        if (threadIdx.x == 0) mosnorm[b * NDP + d] = 0.0f;
        return;
    }
    float ss = 0.0f;
    for (int f = threadIdx.x; f < NF; f += blockDim.x) {
        float v = mos[((size_t)b * NF + f) * NDP + d];
        ss += v * v;
    }
    __shared__ float red[256];
    red[threadIdx.x] = ss;
    __syncthreads();
    for (int s = 128; s > 0; s >>= 1) {
        if (threadIdx.x < s) red[threadIdx.x] += red[threadIdx.x + s];
        __syncthreads();
    }
    if (threadIdx.x == 0)
        mosnorm[b * NDP + d] = sqrtf(Ne[b] / (float)NDIM) / fmaxf(sqrtf(red[0]), EPSN);
}

// K6: v *= mosnorm (in place) over padded [NBF, NDP]; pads stay zero.
__global__ void scale_kernel(float* __restrict__ v, const float* __restrict__ mosnorm) {
    size_t i = (size_t)blockIdx.x * blockDim.x + threadIdx.x;
    if (i >= (size_t)NBF * NDP) return;
    int d   = (int)(i % NDP);
    int row = (int)(i / NDP);
    int b   = row / NF;
    v[i] *= mosnorm[b * NDP + d];
}

// ---------------------------------------------------------------------------
// K7: vout = BN(vin @ WfT + bf) + vin.
// 8 waves / 256 threads per block; block covers 128 rows x all 256 columns, so vin
// is read ONCE for A-fragments (plus once for the residual) per layer.
// WfT (256KB) is staged to LDS with global_load_async_to_lds_b128 (ASYNCcnt path),
// row stride LDSW=272 floats so the two half-wave K-rows hit disjoint LDS banks.
// grid NBF/128 = 500 blocks, dynamic LDS = 278528 B.
__global__ void gemm_layer(const float* __restrict__ vin, const float* __restrict__ WfT,
                           const float* __restrict__ bfp, const float* __restrict__ gamp,
                           const float* __restrict__ betp, const float* __restrict__ rmp,
                           const float* __restrict__ rvp, float* __restrict__ vout) {
    extern __shared__ float shB[];   // [NDP][LDSW]

    // ---- async stage: WfT -> LDS (16B chunks, 64 per thread) ----
    {
        const unsigned ldsBase = (unsigned)(uintptr_t)(void*)shB;  // low 32 bits = LDS offset
        for (int i = threadIdx.x; i < NDP * (NDP / 4); i += 256) {
            const int row = i >> 6;              // NDP/4 = 64 chunks per row
            const int col = (i & 63) << 2;
            const unsigned lds = ldsBase + (unsigned)((row * LDSW + col) * 4);
            const unsigned long long g =
                (unsigned long long)(uintptr_t)(WfT + (size_t)row * NDP + col);
            asm volatile("global_load_async_to_lds_b128 %0, %1, off"
                         :: "v"(lds), "v"(g) : "memory");
        }
        asm volatile("s_wait_asynccnt 0x0" ::: "memory");
        __syncthreads();
    }

    const int lane = threadIdx.x & 31;
    const int wv   = threadIdx.x >> 5;                 // 0..7
    const int m0   = blockIdx.x * 128 + wv * 16;

    v8f acc[16];
#pragma unroll
    for (int t = 0; t < 16; ++t) acc[t] = (v8f){};

#pragma unroll 2
    for (int k = 0; k < NDP; k += 4) {
        const v2f a = load_a2(vin, NDP, m0, k, lane);
        const int kk = k + ((lane >> 4) << 1);
        const float* brow = shB + kk * LDSW + (lane & 15);
#pragma unroll
        for (int t = 0; t < 16; ++t) {
            v2f b;
            b.x = brow[t * 16];
            b.y = brow[LDSW + t * 16];
            acc[t] = wmma4(a, b, acc[t]);
        }
    }

    // epilogue: BN + residual; padded columns come out exactly 0.
    const int mh = (lane >> 4) * 8;
#pragma unroll
    for (int t = 0; t < 16; ++t) {
        const int n = t * 16 + (lane & 15);
        const float g  = gamp[n];
        const float be = betp[n];
        const float rm = rmp[n];
        const float rs = rsqrtf(rvp[n] + BN_EPS);
        const float bb = bfp[n];
#pragma unroll
        for (int r = 0; r < 8; ++r) {
            const size_t m = (size_t)(m0 + mh + r);
            float h = acc[t][r] + bb;
            h = g * (h - rm) * rs + be;
            vout[m * NDP + n] = h + vin[m * NDP + n];
        }
    }
}

// K8: out[b] = sum_f sum_d v[b,f,d] * Wp_pad[d] + bp. grid NB, block 256.
__global__ void out_kernel(const float* __restrict__ v, const float* __restrict__ Wpp,
                           const float* __restrict__ bp, float* __restrict__ out) {
    const int b = blockIdx.x;
    float acc = 0.0f;
    for (int f = 0; f < NF; ++f) {
        const float* row = v + ((size_t)b * NF + f) * NDP;
        for (int d = threadIdx.x; d < NDP; d += blockDim.x)
            acc += row[d] * Wpp[d];
    }
    __shared__ float red[256];
    red[threadIdx.x] = acc;
    __syncthreads();
    for (int s = 128; s > 0; s >>= 1) {
        if (threadIdx.x < s) red[threadIdx.x] += red[threadIdx.x + s];
        __syncthreads();
    }
    if (threadIdx.x == 0) out[b] = red[0] + bp[0];
}

// ---------------------------------------------------------------------------
extern "C" void kernel_launch(void* const* d_in, const int* in_sizes, int n_in,
                              void* d_out, int out_size, void* d_ws, size_t ws_size,
                              hipStream_t stream) {
    (void)in_sizes; (void)n_in; (void)out_size; (void)ws_size;

    const float* dist     = (const float*)d_in[0];   // [B,F,O]
    const int*   qn       = (const int*)  d_in[1];   // [B,O]
    const int*   ao       = (const int*)  d_in[2];   // [B,O]
    const float* Ylm      = (const float*)d_in[3];   // [BO,BO]
    const float* Ne       = (const float*)d_in[4];   // [B]
    const float* coef_emb = (const float*)d_in[5];   // [VOCAB,DIM]
    const float* zeta_emb = (const float*)d_in[6];   // [VOCAB,1]
    const float* Wf       = (const float*)d_in[7];   // [L,DIM,DIM]
    const float* bfv      = (const float*)d_in[8];   // [L,DIM]
    const float* gam      = (const float*)d_in[9];   // [L,DIM]
    const float* bet      = (const float*)d_in[10];  // [L,DIM]
    const float* rmean    = (const float*)d_in[11];  // [L,DIM]
    const float* rvar     = (const float*)d_in[12];  // [L,DIM]
    const float* Wp       = (const float*)d_in[13];  // [NOUT,DIM]
    const float* bp       = (const float*)d_in[14];  // [NOUT]
    float* out = (float*)d_out;

    // carve workspace (256B aligned chunks)
    char* p = (char*)d_ws;
    auto carve = [&](size_t nfloats) {
        float* r = (float*)p;
        p += ((nfloats * sizeof(float) + 255) / 256) * 256;
        return r;
    };
    float* rad      = carve((size_t)NBF * NO);       // 15.36 MB
    float* colscale = carve((size_t)NBO);
    float* coefsN   = carve((size_t)NBO * NDP);      // padded
    float* Wmat     = carve((size_t)NBO * NDP);      // padded
    float* WfT      = carve((size_t)NL * NDP * NDP); // padded transposed weights
    float* bfp      = carve((size_t)NL * NDP);
    float* gamp     = carve((size_t)NL * NDP);
    float* betp     = carve((size_t)NL * NDP);
    float* rmp      = carve((size_t)NL * NDP);
    float* rvp      = carve((size_t)NL * NDP);
    float* Wpp      = carve((size_t)NDP);
    float* mosnorm  = carve((size_t)NB * NDP);
    float* v0       = carve((size_t)NBF * NDP);      // 65.5 MB
    float* v1       = carve((size_t)NBF * NDP);      // 65.5 MB

    // ---- prep (tiny) ----
    prep_wft<<<(NL * NDP * NDP + 255) / 256, 256, 0, stream>>>(Wf, WfT);
    pad_cols<<<(NL * NDP + 255) / 256, 256, 0, stream>>>(bfv,   bfp,  NL);
    pad_cols<<<(NL * NDP + 255) / 256, 256, 0, stream>>>(gam,   gamp, NL);
    pad_cols<<<(NL * NDP + 255) / 256, 256, 0, stream>>>(bet,   betp, NL);
    pad_cols<<<(NL * NDP + 255) / 256, 256, 0, stream>>>(rmean, rmp,  NL);
    pad_cols<<<(NL * NDP + 255) / 256, 256, 0, stream>>>(rvar,  rvp,  NL);
    pad_cols<<<(NDP + 255) / 256, 256, 0, stream>>>(Wp, Wpp, 1);

    // ---- pipeline ----
    radial_kernel<<<dim3(NO, NB), 256, 0, stream>>>(dist, qn, ao, zeta_emb, rad, colscale);
    coef_kernel<<<NB, NDP, 0, stream>>>(ao, coef_emb, coefsN);
    gemm_Ycoef<<<dim3(NDP / 32, NBO / 16), 32, 0, stream>>>(Ylm, coefsN, Wmat);
    gemm_mos<<<dim3(NDP / 32, NF / 16, NB), 32, 0, stream>>>(rad, colscale, Wmat, v0);
    mosnorm_kernel<<<dim3(NDP, NB), 256, 0, stream>>>(v0, Ne, mosnorm);
    {
        size_t total = (size_t)NBF * NDP;
        scale_kernel<<<(unsigned)((total + 255) / 256), 256, 0, stream>>>(v0, mosnorm);
    }
    for (int l = 0; l < NL; ++l) {   // v0 -> v1 -> v0 -> v1
        const float* vin = (l & 1) ? v1 : v0;
        float*       vo  = (l & 1) ? v0 : v1;
        gemm_layer<<<NBF / 128, 256, LAYER_LDS_BYTES, stream>>>(
            vin, WfT + (size_t)l * NDP * NDP, bfp + l * NDP, gamp + l * NDP,
            betp + l * NDP, rmp + l * NDP, rvp + l * NDP, vo);
    }
    out_kernel<<<NB, 256, 0, stream>>>(v1, Wpp, bp, out);
}